// DVGO_61942018343493
// MI455X (gfx1250) — compile-verified
//
#include <hip/hip_runtime.h>
#include <math.h>

// ---------------------------------------------------------------------------
// DVGO volume rendering for MI455X (gfx1250, wave32).
//
// Roofline: density+k0 grids (65 MB) are L2-resident (192 MB L2), so steady
// state is bound by L2 gather bandwidth and VALU/transcendental rate, NOT
// HBM and NOT matrix math. WMMA is structurally inapplicable: the compositing
// einsum is a batched per-ray GEMV (B operand varies per row), so no
// v_wmma/v_swmmac can express it. The gfx1250 paths that matter here:
// wave32 lane-split rays + shuffle prefix-product, b64 z-pair gathers,
// global_prefetch_b8 one sample ahead, raw v_rcp/v_rsq transcendentals.
// ---------------------------------------------------------------------------

namespace {
constexpr int    GRID1  = 160;
constexpr int    GRID2  = GRID1 * GRID1;            // 25600
constexpr long   GRID3  = (long)GRID2 * GRID1;      // 4096000 (k0 channel stride)
constexpr int    NSAMP  = 558;
constexpr int    KSEG   = 4;                        // lanes per ray
constexpr int    SEGLEN = (NSAMP + KSEG - 1) / KSEG;// 140
constexpr float  ACT_SHIFT = -4.595119850134589f;   // log(0.01/0.99)
constexpr float  IDX_SCALE = 79.5f;                 // (GRID-1)/2, align_corners=True
constexpr float  STEP_W    = 0.00625f;              // STEPSIZE * VOXEL_SIZE
constexpr float  LOG2E     = 1.4426950408889634f;
}

// 2-float vector with 4-byte alignment: lets the backend emit global_load_b64
// for z-adjacent corner pairs even though z0 has arbitrary parity.
typedef float v2f_base __attribute__((ext_vector_type(2)));
typedef v2f_base v2f __attribute__((aligned(4)));

__device__ __forceinline__ v2f ld2(const float* __restrict__ p) {
  return *(const v2f*)(p);
}

// z-lerp then xy-weighted sum of 4 preloaded corner pairs
__device__ __forceinline__ float blend4(v2f c00, v2f c01, v2f c10, v2f c11,
                                        float fz, float w00, float w01,
                                        float w10, float w11) {
  float l00 = fmaf(fz, c00.y - c00.x, c00.x);
  float l01 = fmaf(fz, c01.y - c01.x, c01.x);
  float l10 = fmaf(fz, c10.y - c10.x, c10.x);
  float l11 = fmaf(fz, c11.y - c11.x, c11.x);
  return w00 * l00 + w01 * l01 + w10 * l10 + w11 * l11;
}

__device__ __forceinline__ float fast_exp(float x) {   // single v_exp_f32
  return __builtin_amdgcn_exp2f(x * LOG2E);
}

__global__ __launch_bounds__(256)
void dvgo_render(const float* __restrict__ rays_o,
                 const float* __restrict__ rays_d,
                 const float* __restrict__ density,
                 const float* __restrict__ k0,
                 float* __restrict__ out,
                 int n_rays) {
  const int gt  = blockIdx.x * blockDim.x + threadIdx.x;
  const int ray = gt >> 2;        // KSEG == 4
  const int seg = gt & 3;
  if (ray >= n_rays) return;      // grid is sized exactly; never taken

  const float ox = rays_o[ray * 3 + 0];
  const float oy = rays_o[ray * 3 + 1];
  const float oz = rays_o[ray * 3 + 2];
  const float dx = rays_d[ray * 3 + 0];
  const float dy = rays_d[ray * 3 + 1];
  const float dz = rays_d[ray * 3 + 2];

  // ---- ray / bbox intersection (matches reference sample_ray) ----
  const float vx = (dx == 0.0f) ? 1e-6f : dx;
  const float vy = (dy == 0.0f) ? 1e-6f : dy;
  const float vz = (dz == 0.0f) ? 1e-6f : dz;
  const float rax = (1.0f - ox) / vx, rbx = (-1.0f - ox) / vx;
  const float ray_ = (1.0f - oy) / vy, rby = (-1.0f - oy) / vy;
  const float raz = (1.0f - oz) / vz, rbz = (-1.0f - oz) / vz;
  float tmin = fmaxf(fmaxf(fminf(rax, rbx), fminf(ray_, rby)), fminf(raz, rbz));
  float tmax = fminf(fminf(fmaxf(rax, rbx), fmaxf(ray_, rby)), fmaxf(raz, rbz));
  tmin = fminf(fmaxf(tmin, 0.2f), 6.0f);
  tmax = fminf(fmaxf(tmax, 0.2f), 6.0f);
  const bool ray_oob = (tmax < tmin);

  const float dt = STEP_W / sqrtf(dx * dx + dy * dy + dz * dz);

  // per-sample index-space deltas for the next-sample prefetch extrapolation
  const float ddx = dx * dt * IDX_SCALE;
  const float ddy = dy * dt * IDX_SCALE;
  const float ddz = dz * dt * IDX_SCALE;

  // Per-segment state: local transmittance (starts at 1) and local color sum.
  float T  = 1.0f;
  float cr = 0.0f, cg = 0.0f, cb = 0.0f;

  if (!ray_oob) {
    const int s0 = seg * SEGLEN;
    const int s1 = min(NSAMP, s0 + SEGLEN);
    bool inside_seen = false;

    for (int s = s0; s < s1; ++s) {
      const float t  = fmaf(dt, (float)s, tmin);
      const float px = fmaf(dx, t, ox);
      const float py = fmaf(dy, t, oy);
      const float pz = fmaf(dz, t, oz);

      // strict bbox test (identical to reference mask)
      if (px < -1.0f || px > 1.0f || py < -1.0f || py > 1.0f ||
          pz < -1.0f || pz > 1.0f) {
        if (inside_seen) break;   // convex box: once out, stays out
        continue;                 // still before entry
      }
      inside_seen = true;

      // align_corners=True index space; clamp i0 to [0,158] (border-exact)
      const float ix = (px + 1.0f) * IDX_SCALE;
      const float iy = (py + 1.0f) * IDX_SCALE;
      const float iz = (pz + 1.0f) * IDX_SCALE;
      int x0 = (int)floorf(ix);  x0 = min(max(x0, 0), GRID1 - 2);
      int y0 = (int)floorf(iy);  y0 = min(max(y0, 0), GRID1 - 2);
      int z0 = (int)floorf(iz);  z0 = min(max(z0, 0), GRID1 - 2);
      const float fx = ix - (float)x0;
      const float fy = iy - (float)y0;
      const float fz = iz - (float)z0;

      const long base = ((long)x0 * GRID1 + y0) * GRID1 + z0;
      const float* dp = density + base;
      const float* c0 = k0 + base;
      const float* c1 = c0 + GRID3;
      const float* c2 = c1 + GRID3;

      // ---- issue all 16 corner-pair loads up front (max loads in flight) ----
      v2f d00 = ld2(dp),         d01 = ld2(dp + GRID1);
      v2f d10 = ld2(dp + GRID2), d11 = ld2(dp + GRID2 + GRID1);
      v2f r00 = ld2(c0),         r01 = ld2(c0 + GRID1);
      v2f r10 = ld2(c0 + GRID2), r11 = ld2(c0 + GRID2 + GRID1);
      v2f g00 = ld2(c1),         g01 = ld2(c1 + GRID1);
      v2f g10 = ld2(c1 + GRID2), g11 = ld2(c1 + GRID2 + GRID1);
      v2f b00 = ld2(c2),         b01 = ld2(c2 + GRID1);
      v2f b10 = ld2(c2 + GRID2), b11 = ld2(c2 + GRID2 + GRID1);

      // ---- prefetch next sample's 4 plane base lines one iteration ahead ----
      {
        const float jx = ix + ddx, jy = iy + ddy, jz = iz + ddz;
        int u0 = min(max((int)jx, 0), GRID1 - 2);
        int v0 = min(max((int)jy, 0), GRID1 - 2);
        int w0 = min(max((int)jz, 0), GRID1 - 2);
        const long nb = ((long)u0 * GRID1 + v0) * GRID1 + w0;
        __builtin_prefetch(density + nb, 0, 3);
        __builtin_prefetch(k0 + nb, 0, 3);
        __builtin_prefetch(k0 + nb + GRID3, 0, 3);
        __builtin_prefetch(k0 + nb + 2 * GRID3, 0, 3);
      }

      const float gx = 1.0f - fx, gy = 1.0f - fy;
      const float w00 = gx * gy, w01 = gx * fy, w10 = fx * gy, w11 = fx * fy;

      // density -> alpha:  alpha = 1 - (1 + e^x)^(-1/2)   (INTERVAL = 0.5)
      const float dens  = blend4(d00, d01, d10, d11, fz, w00, w01, w10, w11);
      const float rs    = __builtin_amdgcn_rsqf(1.0f + fast_exp(dens + ACT_SHIFT));
      const float alpha = 1.0f - rs;                 // rs == 1 - alpha exactly
      const float w     = alpha * T;
      T *= rs;

      // color: w * sigmoid(trilerp(k0))  — raw v_rcp instead of IEEE divide
      const float vr = blend4(r00, r01, r10, r11, fz, w00, w01, w10, w11);
      const float vg = blend4(g00, g01, g10, g11, fz, w00, w01, w10, w11);
      const float vb = blend4(b00, b01, b10, b11, fz, w00, w01, w10, w11);
      cr += w * __builtin_amdgcn_rcpf(1.0f + fast_exp(-vr));
      cg += w * __builtin_amdgcn_rcpf(1.0f + fast_exp(-vg));
      cb += w * __builtin_amdgcn_rcpf(1.0f + fast_exp(-vb));
    }
  }

  // ---- merge the 4 segments of this ray (aligned 4-lane group, wave32) ----
  // out = sum_l (prod_{j<l} T_j) * c_l ;  final transmittance = prod_l T_l
  float P = 1.0f, R = 0.0f, G = 0.0f, B = 0.0f;
#pragma unroll
  for (int l = 0; l < KSEG; ++l) {
    const float Tl = __shfl(T,  l, KSEG);
    R += P * __shfl(cr, l, KSEG);
    G += P * __shfl(cg, l, KSEG);
    B += P * __shfl(cb, l, KSEG);
    P *= Tl;
  }

  if (seg == 0) {
    out[ray * 3 + 0] = R + P;   // + alphainv_cum_last * BG (BG == 1)
    out[ray * 3 + 1] = G + P;
    out[ray * 3 + 2] = B + P;
  }
}

extern "C" void kernel_launch(void* const* d_in, const int* in_sizes, int n_in,
                              void* d_out, int out_size, void* d_ws, size_t ws_size,
                              hipStream_t stream) {
  (void)n_in; (void)out_size; (void)d_ws; (void)ws_size;
  const float* rays_o  = (const float*)d_in[0];
  const float* rays_d  = (const float*)d_in[1];
  const float* density = (const float*)d_in[2];
  const float* k0      = (const float*)d_in[3];
  float* out = (float*)d_out;

  const int n_rays  = in_sizes[0] / 3;          // 16384
  const int threads = n_rays * KSEG;            // 65536
  dim3 block(256);                              // 8 wave32 per block
  dim3 grid((threads + 255) / 256);             // 256 blocks -> 2048 waves
  hipLaunchKernelGGL(dvgo_render, grid, block, 0, stream,
                     rays_o, rays_d, density, k0, out, n_rays);
}